// acm_eachloss_hnode_prompt_layer_feature_weighted_sum_21534966022306
// MI455X (gfx1250) — compile-verified
//
#include <hip/hip_runtime.h>
#include <stdint.h>

// ---------------------------------------------------------------------------
// acm_eachloss_hnode_prompt_layer_feature_weighted_sum  (graph scatter-sum)
//   res  = segment_sum((X*w)[src], dst)          [N,128]
//   res0 = segment_sum(where(e==0,(X*w)[src],0), dst)
//   out  = concat(res, res0, res0, res0, res0)
// Memory-bound (AI ~0.3 FLOP/B): gfx1250 async global->LDS gather pipeline
// (ASYNCcnt ring, depth 8), B128 vector accesses, and explicit native
// global_atomic_add_f32 (no-return, saddr form) for the scatter.
// ---------------------------------------------------------------------------

#define TPB 256
#define WAVES_PER_BLOCK (TPB / 32)
#define DEPTH 8           // async ring depth per wave (ASYNCcnt is 6 bits)
#define ROW_F 128         // D
#define ROW_BYTES 512     // D * sizeof(float)

// gfx1250 async gather: each lane moves 16B global -> LDS, 32 lanes = one row.
// GVS addressing: lds_dst(v32), voffset(v32), base(s64). Tracked by ASYNCcnt.
__device__ __forceinline__ void async_row_load(unsigned lds_off, unsigned voff,
                                               uint64_t gbase) {
  asm volatile("global_load_async_to_lds_b128 %0, %1, %2"
               :: "v"(lds_off), "v"(voff), "s"(gbase) : "memory");
}
// Consume-oldest wait: with DEPTH(=8) ops in flight, asynccnt<=7 means the
// oldest async load has landed in LDS (async loads complete in order).
__device__ __forceinline__ void wait_async_oldest() {
  asm volatile("s_wait_asynccnt 7" ::: "memory");
}
// Protect ring-slot reuse: the ds_load of the slot must finish before the
// refill async write targets the same LDS bytes.
__device__ __forceinline__ void wait_ds0() {
  asm volatile("s_wait_dscnt 0" ::: "memory");
}
// Native no-return f32 atomics: 4 contiguous floats, one voffset VGPR +
// offset immediates. Fire-and-forget (STOREcnt), EXEC-respecting.
__device__ __forceinline__ void atomic_add_row4(unsigned voff, const float4& m,
                                                uint64_t base) {
  asm volatile(
      "global_atomic_add_f32 %0, %1, %5\n\t"
      "global_atomic_add_f32 %0, %2, %5 offset:4\n\t"
      "global_atomic_add_f32 %0, %3, %5 offset:8\n\t"
      "global_atomic_add_f32 %0, %4, %5 offset:12"
      :: "v"(voff), "v"(m.x), "v"(m.y), "v"(m.z), "v"(m.w), "s"(base)
      : "memory");
}

// Low 32 bits of a generic pointer to a __shared__ object = wave-relative LDS
// byte offset (AS3->generic addrspacecast is {aperture_hi32, lds_offset}).
__device__ __forceinline__ unsigned lds_byte_offset(const void* p) {
  return (unsigned)(uintptr_t)p;
}

// ---- hot kernel first so the disasm snippet shows it --------------------
__global__ void __launch_bounds__(TPB) edge_scatter_kernel(
    const float* __restrict__ gemb,    // [N,128] node features
    const float* __restrict__ weight,  // [128]
    const int* __restrict__ src, const int* __restrict__ dst,
    const int* __restrict__ efeat,
    float* __restrict__ res, float* __restrict__ res0,
    int nEdges, int nWaves) {
  // Per-wave DEPTH-deep ring of 512B rows + one shared trash row for dummy
  // refills (keeps ASYNCcnt bookkeeping uniform through the loop tail).
  // LDS: 8 waves * 8 * 512B + 512B = 33 KB of 320 KB -> occupancy unaffected.
  __shared__ __align__(16) float smem[WAVES_PER_BLOCK * DEPTH * ROW_F + ROW_F];

  const int lane = threadIdx.x & 31;
  const int wib  = threadIdx.x >> 5;
  const int wid  = blockIdx.x * WAVES_PER_BLOCK + wib;

  const float4 w4 = *(const float4*)(weight + lane * 4);

  const unsigned laneByte = (unsigned)lane * 16u;
  const unsigned ldsBase  = lds_byte_offset(&smem[wib * DEPTH * ROW_F]);
  const unsigned ldsTrash = lds_byte_offset(&smem[WAVES_PER_BLOCK * DEPTH * ROW_F]);
  const uint64_t gbase    = (uint64_t)(uintptr_t)gemb;
  const uint64_t resBase  = (uint64_t)(uintptr_t)res;
  const uint64_t res0Base = (uint64_t)(uintptr_t)res0;

  // Prologue: fill the ring (dummy loads of row 0 past the edge list end).
  for (int p = 0; p < DEPTH; ++p) {
    const long long ee = (long long)wid + (long long)p * nWaves;
    unsigned voff, ldst;
    if (ee < nEdges) {
      voff = (unsigned)src[ee] * (unsigned)ROW_BYTES + laneByte;
      ldst = ldsBase + (unsigned)p * (unsigned)ROW_BYTES + laneByte;
    } else {
      voff = laneByte;
      ldst = ldsTrash + laneByte;
    }
    async_row_load(ldst, voff, gbase);
  }

  long long k = 0;
  for (long long e = wid; e < nEdges; e += nWaves, ++k) {
    wait_async_oldest();  // oldest ring slot is resident in LDS
    const int slot = (int)(k & (DEPTH - 1));
    const float4 v =
        *(const float4*)&smem[(wib * DEPTH + slot) * ROW_F + lane * 4];
    wait_ds0();  // slot read complete -> safe to overwrite

    // Refill this slot for edge e + DEPTH*nWaves (or a dummy), keeping
    // exactly DEPTH async ops in flight so the constant wait stays valid.
    const long long en = e + (long long)DEPTH * nWaves;
    unsigned voff, ldst;
    if (en < nEdges) {
      voff = (unsigned)src[en] * (unsigned)ROW_BYTES + laneByte;
      ldst = ldsBase + (unsigned)slot * (unsigned)ROW_BYTES + laneByte;
    } else {
      voff = laneByte;
      ldst = ldsTrash + laneByte;
    }
    async_row_load(ldst, voff, gbase);

    const int d = dst[e];
    const int f = efeat[e];
    float4 m;
    m.x = v.x * w4.x; m.y = v.y * w4.y;
    m.z = v.z * w4.z; m.w = v.w * w4.w;

    // Scatter: same row-offset structure as the gather (d*512 + lane*16).
    const unsigned roff = (unsigned)d * (unsigned)ROW_BYTES + laneByte;
    atomic_add_row4(roff, m, resBase);
    if (f == 0) {
      atomic_add_row4(roff, m, res0Base);
    }
  }
}

__global__ void __launch_bounds__(TPB) zero_f4_kernel(float4* __restrict__ p,
                                                      long long n4) {
  long long i = (long long)blockIdx.x * blockDim.x + threadIdx.x;
  const long long stride = (long long)gridDim.x * blockDim.x;
  const float4 z = make_float4(0.f, 0.f, 0.f, 0.f);
  for (; i < n4; i += stride) p[i] = z;
}

// out slots 2..4 = res0 (B128 read once, three B128 stores).
__global__ void __launch_bounds__(TPB) replicate3_kernel(
    const float4* __restrict__ src1, float4* __restrict__ dsts, long long n4) {
  long long i = (long long)blockIdx.x * blockDim.x + threadIdx.x;
  const long long stride = (long long)gridDim.x * blockDim.x;
  for (; i < n4; i += stride) {
    const float4 v = src1[i];
    dsts[i] = v;
    dsts[i + n4] = v;
    dsts[i + 2 * n4] = v;
  }
}

extern "C" void kernel_launch(void* const* d_in, const int* in_sizes, int n_in,
                              void* d_out, int out_size, void* d_ws,
                              size_t ws_size, hipStream_t stream) {
  const float* gemb = (const float*)d_in[0];
  const float* w    = (const float*)d_in[1];
  const int* src    = (const int*)d_in[2];
  const int* dst    = (const int*)d_in[3];
  const int* ef     = (const int*)d_in[4];

  const int N = in_sizes[0] / ROW_F;   // 50000
  const int E = in_sizes[2];           // 800000

  float* out  = (float*)d_out;
  float* res  = out;                                 // slot 0
  float* res0 = out + (size_t)N * ROW_F;             // slot 1

  // 1) zero accumulators (harness poisons d_out with 0xAA)
  const long long nz4 = (long long)2 * N * ROW_F / 4;
  zero_f4_kernel<<<2048, TPB, 0, stream>>>((float4*)out, nz4);

  // 2) pipelined gather -> scale -> atomic scatter
  const int blocks = 1024;  // 8192 waves, ~98 edges each through the ring
  edge_scatter_kernel<<<blocks, TPB, 0, stream>>>(gemb, w, src, dst, ef, res,
                                                  res0, E, blocks * WAVES_PER_BLOCK);

  // 3) slots 2..4 = res0
  const long long n4 = (long long)N * ROW_F / 4;
  replicate3_kernel<<<2048, TPB, 0, stream>>>(
      (const float4*)res0, (float4*)(out + (size_t)2 * N * ROW_F), n4);
}